// DMoNModel_76914274337337
// MI455X (gfx1250) — compile-verified
//
#include <hip/hip_runtime.h>
#include <hip/hip_bf16.h>

typedef float v2f __attribute__((ext_vector_type(2)));
typedef float v8f __attribute__((ext_vector_type(8)));

#define D_FEAT 128

// ---------------------------------------------------------------------------
// C[M,N] = A[M,K] @ B[K,N] + bias[N]
// A row-major (ld=K), B row-major (ld=N). One wave32 computes one 16x16 tile
// using V_WMMA_F32_16X16X4_F32, stepping K by 4.
// Requires M % 16 == 0, N % 16 == 0, K % 4 == 0.
// ---------------------------------------------------------------------------
__global__ void gemm_bias_wmma(const float* __restrict__ A,
                               const float* __restrict__ B,
                               const float* __restrict__ bias,
                               float* __restrict__ C,
                               int M, int N, int K) {
    const int wave = threadIdx.x >> 5;
    const int lane = threadIdx.x & 31;
    const int wavesPerBlock = blockDim.x >> 5;
    const int colTiles = N >> 4;

    int flat = blockIdx.x * wavesPerBlock + wave;      // flat tile id
    int rowTile = flat / colTiles;
    int colTile = flat - rowTile * colTiles;
    if (rowTile * 16 >= M) return;

    const int half = lane >> 4;      // 0: lanes 0-15, 1: lanes 16-31
    const int l16  = lane & 15;
    const int arow = rowTile * 16 + l16;               // A row for this lane
    const int col  = colTile * 16 + l16;               // B/C column for this lane

    const float* aPtr = A + (size_t)arow * K;

    v8f acc = {};
    for (int k0 = 0; k0 < K; k0 += 4) {
        // fp32 A 16x4 layout: VGPR r holds K = k0 + 2*half + r
        v2f a;
        a.x = aPtr[k0 + 2 * half + 0];
        a.y = aPtr[k0 + 2 * half + 1];
        // fp32 B 4x16 layout: VGPR r holds K = k0 + 2*half + r, N = l16
        v2f b;
        b.x = B[(size_t)(k0 + 2 * half + 0) * N + col];
        b.y = B[(size_t)(k0 + 2 * half + 1) * N + col];
        acc = __builtin_amdgcn_wmma_f32_16x16x4_f32(
            /*neg_a=*/false, a, /*neg_b=*/false, b,
            /*c_mod=*/(short)0, acc, /*reuse_a=*/false, /*reuse_b=*/false);
    }

    const float bv = bias[col];
    // C/D layout: VGPR v, lane -> M = rowTile*16 + v + 8*half, N = col
#pragma unroll
    for (int v = 0; v < 8; ++v) {
        C[(size_t)(rowTile * 16 + v + 8 * half) * N + col] = acc[v] + bv;
    }
}

// ---------------------------------------------------------------------------
// P[dst] += w_e * T[src]  (segment-sum over edges). One wave per edge;
// each lane handles 4 consecutive floats (float4 load, 4 hw f32 atomics).
// ---------------------------------------------------------------------------
__global__ void spmm_atomic(const int* __restrict__ ei,
                            const float* __restrict__ ew,
                            const float* __restrict__ T,
                            float* __restrict__ P,
                            int E) {
    const int wave = threadIdx.x >> 5;
    const int lane = threadIdx.x & 31;
    const int e = blockIdx.x * (blockDim.x >> 5) + wave;
    if (e >= E) return;

    const int dst = ei[e];
    const int src = ei[E + e];
    const float w = ew[e];

    const float4 v = *(const float4*)(T + (size_t)src * D_FEAT + lane * 4);
    float* p = P + (size_t)dst * D_FEAT + lane * 4;
    unsafeAtomicAdd(p + 0, w * v.x);
    unsafeAtomicAdd(p + 1, w * v.y);
    unsafeAtomicAdd(p + 2, w * v.z);
    unsafeAtomicAdd(p + 3, w * v.w);
}

// ---------------------------------------------------------------------------
// H = selu(skip[col] * T + P), elementwise over [M, 128], float4 vectorized.
// ---------------------------------------------------------------------------
__global__ void selu_skip(const float* __restrict__ T,
                          const float* __restrict__ P,
                          const float* __restrict__ skip,
                          float* __restrict__ H,
                          int n4) {
    const int i = blockIdx.x * blockDim.x + threadIdx.x;
    if (i >= n4) return;
    const float SCALE = 1.0507009873554805f;
    const float ALPHA = 1.6732632423543772f;

    const float4 t = ((const float4*)T)[i];
    const float4 p = ((const float4*)P)[i];
    const int c = (i * 4) & (D_FEAT - 1);
    const float4 s = *(const float4*)(skip + c);

    float4 r;
    {
        float x = s.x * t.x + p.x;
        r.x = x > 0.f ? SCALE * x : SCALE * ALPHA * (expf(x) - 1.f);
    }
    {
        float x = s.y * t.y + p.y;
        r.y = x > 0.f ? SCALE * x : SCALE * ALPHA * (expf(x) - 1.f);
    }
    {
        float x = s.z * t.z + p.z;
        r.z = x > 0.f ? SCALE * x : SCALE * ALPHA * (expf(x) - 1.f);
    }
    {
        float x = s.w * t.w + p.w;
        r.w = x > 0.f ? SCALE * x : SCALE * ALPHA * (expf(x) - 1.f);
    }
    ((float4*)H)[i] = r;
}

// ---------------------------------------------------------------------------
// out = softmax(H @ Wa + ba) with Kc == 16: one wave computes the 16x16
// logits tile via WMMA, then does the row softmax in registers using
// __shfl_xor across the 16-lane N group.
// ---------------------------------------------------------------------------
__global__ void assign_softmax(const float* __restrict__ H,
                               const float* __restrict__ Wa,
                               const float* __restrict__ ba,
                               float* __restrict__ out,
                               int M, int K) {
    const int wave = threadIdx.x >> 5;
    const int lane = threadIdx.x & 31;
    const int rowTile = blockIdx.x * (blockDim.x >> 5) + wave;
    if (rowTile * 16 >= M) return;

    const int half = lane >> 4;
    const int l16  = lane & 15;
    const int arow = rowTile * 16 + l16;
    const float* aPtr = H + (size_t)arow * K;

    v8f acc = {};
    for (int k0 = 0; k0 < K; k0 += 4) {
        v2f a;
        a.x = aPtr[k0 + 2 * half + 0];
        a.y = aPtr[k0 + 2 * half + 1];
        v2f b;
        b.x = Wa[(size_t)(k0 + 2 * half + 0) * 16 + l16];
        b.y = Wa[(size_t)(k0 + 2 * half + 1) * 16 + l16];
        acc = __builtin_amdgcn_wmma_f32_16x16x4_f32(
            false, a, false, b, (short)0, acc, false, false);
    }

    const float bv = ba[l16];
#pragma unroll
    for (int v = 0; v < 8; ++v) {
        float x = acc[v] + bv;                 // logit(row = v + 8*half, col = l16)
        // row-max across the 16-lane N group
        float m = x;
#pragma unroll
        for (int off = 1; off < 16; off <<= 1)
            m = fmaxf(m, __shfl_xor(m, off, 32));
        float e = expf(x - m);
        float sum = e;
#pragma unroll
        for (int off = 1; off < 16; off <<= 1)
            sum += __shfl_xor(sum, off, 32);
        out[(size_t)(rowTile * 16 + v + 8 * half) * 16 + l16] = e / sum;
    }
}

// ---------------------------------------------------------------------------
// Host launcher
// ---------------------------------------------------------------------------
extern "C" void kernel_launch(void* const* d_in, const int* in_sizes, int n_in,
                              void* d_out, int out_size, void* d_ws, size_t ws_size,
                              hipStream_t stream) {
    const float* x  = (const float*)d_in[0];
    const int*   ei = (const int*)d_in[1];
    const float* ew = (const float*)d_in[2];
    const float* W1 = (const float*)d_in[3];
    const float* b1 = (const float*)d_in[4];
    const float* s1 = (const float*)d_in[5];
    const float* W2 = (const float*)d_in[6];
    const float* b2 = (const float*)d_in[7];
    const float* s2 = (const float*)d_in[8];
    const float* Wa = (const float*)d_in[9];
    const float* ba = (const float*)d_in[10];
    float* out = (float*)d_out;

    const int D = D_FEAT;
    const int M = in_sizes[0] / D;      // 50000 nodes
    const int E = in_sizes[1] / 2;      // 1.6M edges

    float* T = (float*)d_ws;            // transformed = h @ W + b
    float* P = T + (size_t)M * D;       // propagated (segment sum)
    float* H = P + (size_t)M * D;       // layer output
    const size_t featBytes = (size_t)M * D * sizeof(float);

    const dim3 blk(256);
    const int rowTiles = (M + 15) / 16;
    const dim3 gemmGrid((rowTiles * (D / 16) + 7) / 8);
    const dim3 spmmGrid((E + 7) / 8);
    const int n4 = (M * D) / 4;
    const dim3 eltGrid((n4 + 255) / 256);
    const dim3 asgGrid((rowTiles + 7) / 8);

    // ---- Layer 1 ----
    gemm_bias_wmma<<<gemmGrid, blk, 0, stream>>>(x, W1, b1, T, M, D, D);
    hipMemsetAsync(P, 0, featBytes, stream);
    spmm_atomic<<<spmmGrid, blk, 0, stream>>>(ei, ew, T, P, E);
    selu_skip<<<eltGrid, blk, 0, stream>>>(T, P, s1, H, n4);

    // ---- Layer 2 ----
    gemm_bias_wmma<<<gemmGrid, blk, 0, stream>>>(H, W2, b2, T, M, D, D);
    hipMemsetAsync(P, 0, featBytes, stream);
    spmm_atomic<<<spmmGrid, blk, 0, stream>>>(ei, ew, T, P, E);
    selu_skip<<<eltGrid, blk, 0, stream>>>(T, P, s2, H, n4);

    // ---- Assignment head: softmax(H @ Wa + ba) ----
    assign_softmax<<<asgGrid, blk, 0, stream>>>(H, Wa, ba, out, M, D);
}